// EdgePooling_88553635709188
// MI455X (gfx1250) — compile-verified
//
#include <hip/hip_runtime.h>
#include <hip/hip_bf16.h>
#include <cstddef>

// ---------------------------------------------------------------------------
// EdgePooling edge scoring, factored form:
//   s1[n] = x[n,:] . W[0:128]      (per node, computed once via WMMA GEMV)
//   s2[n] = x[n,:] . W[128:256]
//   e[i]  = sigmoid(s1[src] + s2[dst] + b) + 0.3
// Outputs (flat, in return order): x copy | edge_index (as f32) | e
// ---------------------------------------------------------------------------

typedef __attribute__((ext_vector_type(2))) float v2f;
typedef __attribute__((ext_vector_type(8))) float v8f;

#define C_DIM 128

// One wave32 computes a 16-node tile of (s1, s2) using V_WMMA_F32_16X16X4_F32.
// A (16x4 f32): lanes 0-15 hold K=k0,k0+1 (VGPR0,1); lanes 16-31 hold K=k0+2,k0+3.
// B (4x16 f32): lane's N = lane&15; column 0 carries W[0:128], column 1 carries
//               W[128:256], columns 2..15 read a zero-padded LDS region, so the
//               inner loop has no per-lane selects (EXEC stays all-ones).
// C/D (16x16 f32, 8 VGPRs): VGPR v: lanes 0-15 -> (M=v, N=lane),
//                                    lanes 16-31 -> (M=v+8, N=lane-16).
__global__ __launch_bounds__(256) void node_scores_wmma_kernel(
    const float* __restrict__ x, const float* __restrict__ W,
    float* __restrict__ s1, float* __restrict__ s2, int n_nodes) {
  __shared__ __align__(16) float Wlds[384];   // [0:256)=W, [256:384)=zeros
  __shared__ __align__(16) float Stage[256];  // 8 waves x {s1[16], s2[16]}

  const int tid = threadIdx.x;  // 0..255 (8 waves)
  Wlds[tid] = W[tid];
  if (tid < 128) Wlds[256 + tid] = 0.0f;
  __syncthreads();

  const int wave = tid >> 5;
  const int lane = tid & 31;
  const int half = lane >> 4;  // 0: K pair {k0,k0+1}, 1: {k0+2,k0+3}
  const int l16  = lane & 15;  // A: row M; B: column N

  const int node_base = blockIdx.x * 128 + wave * 16;
  const int row = node_base + l16;
  const int row_c = row < (n_nodes - 1) ? row : (n_nodes - 1);  // clamp loads
  const float* __restrict__ xrow = x + (size_t)row_c * C_DIM + half * 2;

  // Per-lane, loop-invariant B base: N==0 -> W[0:], N==1 -> W[128:], else pad.
  const float* __restrict__ wrow =
      &Wlds[((l16 == 0) ? 0 : ((l16 == 1) ? 128 : 256)) + half * 2];

  v8f c = {};
#pragma unroll 4
  for (int k0 = 0; k0 < C_DIM; k0 += 4) {
    const v2f a  = *(const v2f*)(xrow + k0);  // global_load_b64
    const v2f bm = *(const v2f*)(wrow + k0);  // ds_load_b64
    // (neg_a, A, neg_b, B, c_mod, C, reuse_a, reuse_b)
    c = __builtin_amdgcn_wmma_f32_16x16x4_f32(false, a, false, bm, (short)0, c,
                                              false, false);
  }

  // Epilogue: result columns N=0 (s1) / N=1 (s2) live in lanes 0,1,16,17.
  // Each of those 4 lanes owns 8 contiguous stage slots -> two ds_store_b128,
  // then every lane does exactly one predicated contiguous global store.
  float* st = &Stage[wave * 32];
  if (l16 < 2) {
    const int base = l16 * 16 + half * 8;  // 16B-aligned
    float4 lo = make_float4(c[0], c[1], c[2], c[3]);
    float4 hi = make_float4(c[4], c[5], c[6], c[7]);
    *(float4*)(&st[base])     = lo;
    *(float4*)(&st[base + 4]) = hi;
  }
  asm volatile("s_wait_dscnt 0x0" ::: "memory");  // wave-local LDS ordering
  if (row < n_nodes) {
    float* __restrict__ dst = half ? s2 : s1;  // lanes 0-15: s1, 16-31: s2
    dst[row] = st[lane];
  }
}

// Per-edge: two scalar gathers from L2-resident tables, sigmoid, +0.3.
// Also emits edge_index cast to the output dtype (f32).
__global__ __launch_bounds__(256) void edge_score_kernel(
    const long long* __restrict__ ei, const float* __restrict__ s1,
    const float* __restrict__ s2, const float* __restrict__ bptr,
    float* __restrict__ ei_out, float* __restrict__ e_out, int n_edges) {
  const int i = blockIdx.x * blockDim.x + threadIdx.x;
  if (i >= n_edges) return;
  const long long sn = ei[i];
  const long long dn = ei[n_edges + i];
  const float s = s1[sn] + s2[dn] + bptr[0];
  e_out[i] = 1.0f / (1.0f + __expf(-s)) + 0.3f;
  ei_out[i] = (float)sn;
  ei_out[n_edges + i] = (float)dn;
}

// Straight float4 streaming copy of x into the output slab.
__global__ __launch_bounds__(256) void copy_x_kernel(
    const float4* __restrict__ in, float4* __restrict__ out, int n4) {
  const int i = blockIdx.x * blockDim.x + threadIdx.x;
  if (i < n4) out[i] = in[i];
}

extern "C" void kernel_launch(void* const* d_in, const int* in_sizes, int n_in,
                              void* d_out, int out_size, void* d_ws,
                              size_t ws_size, hipStream_t stream) {
  const float* x = (const float*)d_in[0];
  const long long* ei = (const long long*)d_in[1];  // int64 edge_index
  const float* W = (const float*)d_in[2];
  const float* b = (const float*)d_in[3];

  const int n_x = in_sizes[0];      // N_NODES * 128
  const int n_nodes = n_x / C_DIM;  // 100000
  const int n_ei = in_sizes[1];     // 2 * N_EDGES
  const int n_edges = n_ei / 2;     // 640000

  float* out = (float*)d_out;
  float* out_x = out;               // [n_x]
  float* out_ei = out + n_x;        // [n_ei]
  float* out_e = out + n_x + n_ei;  // [n_edges]

  float* s1 = (float*)d_ws;         // [n_nodes]
  float* s2 = s1 + n_nodes;         // [n_nodes]

  // 1) Node score tables via WMMA GEMV (reads x exactly once).
  {
    const int blocks = (n_nodes + 127) / 128;  // 128 nodes / block (8 waves)
    node_scores_wmma_kernel<<<blocks, 256, 0, stream>>>(x, W, s1, s2, n_nodes);
  }
  // 2) x passthrough copy (bandwidth-bound float4 stream).
  {
    const int n4 = n_x / 4;
    const int blocks = (n4 + 255) / 256;
    copy_x_kernel<<<blocks, 256, 0, stream>>>((const float4*)x, (float4*)out_x,
                                              n4);
  }
  // 3) Edge scores + edge_index passthrough.
  {
    const int blocks = (n_edges + 255) / 256;
    edge_score_kernel<<<blocks, 256, 0, stream>>>(ei, s1, s2, b, out_ei, out_e,
                                                  n_edges);
  }
}